// GNN_30296699306730
// MI455X (gfx1250) — compile-verified
//
#include <hip/hip_runtime.h>

// GNN message-passing fused kernel for MI455X (gfx1250).
//  * x-slices staged into LDS with gfx1250 async global->LDS DMA (ASYNCcnt).
//  * Edge phase: scalar VALU message MLP + gate, ds_add_f32 LDS scatter-add
//    (gather/scatter bound; no WMMA shape applies to K<=4 with per-edge
//    nonlinearities at 32 lanes/VALU-instr).
//  * Node phase: 4-layer MLP chained on V_WMMA_F32_16X16X4_F32 using the
//    weights-in-A / activations-in-B-C-D trick: D's row0/row1+row8/row9
//    layout is bit-identical to the next WMMA's B layout (K0,K1 valid,
//    K2,K3 zero), so layers chain with zero cross-lane data movement.
//  * Sigmoid uses hardware v_rcp_f32 instead of the IEEE div sequence.

typedef __attribute__((ext_vector_type(2))) float v2f;
typedef __attribute__((ext_vector_type(8))) float v8f;

namespace {

constexpr int kB = 4, kT = 64, kN = 1000, kE = 32000, kF = 2;
constexpr int kSlices  = 2;                     // (b,t) slices per workgroup
constexpr int kThreads = 512;                   // 16 wave32 waves
constexpr int kWaves   = kThreads / 32;
constexpr int kSliceF  = kN * kF;               // 2000 floats per slice
constexpr int kBlocks  = (kB * kT) / kSlices;   // 128 workgroups
constexpr int kNodeSlices = kSlices * kN;       // 2000 node-slices per block

// LDS byte offset of a shared-memory object (generic -> AS(3) -> int).
__device__ __forceinline__ unsigned lds_byte_off(const void* p) {
  return (unsigned)(unsigned long long)(__attribute__((address_space(3))) const void*)p;
}

// gfx1250 async DMA: 16B global -> LDS, tracked by ASYNCcnt (no VGPR data).
__device__ __forceinline__ void async_load_b128(unsigned lds_byte, const void* gaddr) {
  asm volatile("global_load_async_to_lds_b128 %0, %1, off"
               :: "v"(lds_byte), "v"(gaddr) : "memory");
}

__device__ __forceinline__ void wait_async_zero() {
  asm volatile("s_wait_asynccnt 0" ::: "memory");
}

__device__ __forceinline__ float relu(float v)  { return fmaxf(v, 0.0f); }
// sigmoid via hardware transcendentals: v_exp_f32 + v_rcp_f32 (~1 ulp).
__device__ __forceinline__ float sigm(float z) {
  return __builtin_amdgcn_rcpf(1.0f + __expf(-z));
}

// D = A(16x4) x B(4x16) + C, f32.
__device__ __forceinline__ v8f wmma_k4(v2f a, v2f b, v8f c) {
  return __builtin_amdgcn_wmma_f32_16x16x4_f32(false, a, false, b, (short)0, c,
                                               false, false);
}

__global__ __launch_bounds__(kThreads)
void gnn_fused(const float* __restrict__ x,       // (B,T,N,F)
               const long long* __restrict__ ei,  // (2,E) int64
               const float* __restrict__ mw1, const float* __restrict__ mb1,
               const float* __restrict__ mw2, const float* __restrict__ mb2,
               const float* __restrict__ gw,  const float* __restrict__ gb,
               const float* __restrict__ uw1, const float* __restrict__ ub1,
               const float* __restrict__ uw2, const float* __restrict__ ub2,
               const float* __restrict__ fw1, const float* __restrict__ fb1,
               const float* __restrict__ fw2, const float* __restrict__ fb2,
               float* __restrict__ out)            // (B,T,N,1)
{
  __shared__ __align__(16) float s_x[kNodeSlices * kF];    // 16 KB
  __shared__ __align__(16) float s_agg[kNodeSlices * kF];  // 16 KB

  const int tid = threadIdx.x;

  // ---- Stage x slices: one contiguous 16 KB region via async DMA.
  const float* gx = x + (size_t)blockIdx.x * (kSlices * kSliceF);
  const unsigned lds_x0 = lds_byte_off(s_x);
  constexpr int kVecs = (kSlices * kSliceF) / 4;  // b128 transfers
  for (int i = tid; i < kVecs; i += kThreads) {
    async_load_b128(lds_x0 + (unsigned)i * 16u, gx + i * 4);
  }

  // ---- Zero the aggregation buffer while the DMA is in flight.
  for (int i = tid; i < kSlices * kSliceF; i += kThreads) s_agg[i] = 0.0f;

  // ---- Hoist the 37 scalar weights into registers (uniform loads).
  const float w10 = mw1[0], w11 = mw1[1], w12 = mw1[2], w13 = mw1[3], b1 = mb1[0];
  const float w20 = mw2[0], w21 = mw2[1], b20 = mb2[0], b21 = mb2[1];
  const float g0 = gw[0], g1 = gw[1], gbb = gb[0];
  const float u00 = uw1[0], u01 = uw1[1], u10 = uw1[2], u11 = uw1[3];  // W1[k][m]
  const float u20 = uw1[4], u21 = uw1[5], u30 = uw1[6], u31 = uw1[7];
  const float ubb0 = ub1[0], ubb1 = ub1[1];
  const float v00 = uw2[0], v01 = uw2[1], v10 = uw2[2], v11 = uw2[3];  // W2[k][m]
  const float vb0 = ub2[0], vb1 = ub2[1];
  const float f00 = fw1[0], f01 = fw1[1], f10 = fw1[2], f11 = fw1[3];  // Wf1[k][m]
  const float fb10 = fb1[0], fb11 = fb1[1];
  const float q0 = fw2[0], q1 = fw2[1], qb = fb2[0];

  wait_async_zero();   // this wave's DMA done
  __syncthreads();     // all waves' DMA visible in LDS

  // ---- Edge phase: gather from LDS, 3-layer message+gate, LDS scatter-add.
  const long long* src_p = ei;
  const long long* dst_p = ei + kE;
  for (int e = tid; e < kE; e += kThreads) {
    const int sn = (int)src_p[e];
    const int dn = (int)dst_p[e];
#pragma unroll
    for (int s = 0; s < kSlices; ++s) {
      const float* xs = s_x + s * kSliceF;
      const float xi0 = xs[2 * dn], xi1 = xs[2 * dn + 1];
      const float xj0 = xs[2 * sn], xj1 = xs[2 * sn + 1];
      const float m1 = relu(fmaf(w10, xi0, fmaf(w11, xi1, fmaf(w12, xj0, fmaf(w13, xj1, b1)))));
      const float m0 = relu(fmaf(w20, m1, b20));
      const float m1b = relu(fmaf(w21, m1, b21));
      const float gate = sigm(fmaf(g0, m0, fmaf(g1, m1b, gbb)));
      float* ag = s_agg + s * kSliceF + 2 * dn;
      atomicAdd(ag,     gate * m0);   // ds_add_f32
      atomicAdd(ag + 1, gate * m1b);  // ds_add_f32
    }
  }
  __syncthreads();

  // ---- Node phase on WMMA: weights in A, 16 nodes as B/C/D columns.
  const int lane = tid & 31;
  const int m    = lane & 15;        // A-matrix row / node-within-group
  const bool lo  = lane < 16;        // lanes 0-15 carry K0,K1 (A) / rows 0-7 (D)

  // A1 = up_w1^T (2x4): lane<16 holds A[m][0..1], lane>=16 holds A[m][2..3].
  v2f a1; a1.x = (m == 0) ? (lo ? u00 : u20) : (m == 1) ? (lo ? u01 : u21) : 0.f;
          a1.y = (m == 0) ? (lo ? u10 : u30) : (m == 1) ? (lo ? u11 : u31) : 0.f;
  // A2 = up_w2^T (2x2): K2,K3 zero -> high lanes zero.
  v2f a2; a2.x = (lo && m == 0) ? v00 : (lo && m == 1) ? v01 : 0.f;
          a2.y = (lo && m == 0) ? v10 : (lo && m == 1) ? v11 : 0.f;
  // A3 = fc_w1^T (2x2).
  v2f a3; a3.x = (lo && m == 0) ? f00 : (lo && m == 1) ? f01 : 0.f;
          a3.y = (lo && m == 0) ? f10 : (lo && m == 1) ? f11 : 0.f;
  // A4 = fc_w2^T (1x2).
  v2f a4; a4.x = (lo && m == 0) ? q0 : 0.f;
          a4.y = (lo && m == 0) ? q1 : 0.f;

  constexpr int kPerIter = kWaves * 16;  // node-slices per block-iteration
  constexpr int kIters = (kNodeSlices + kPerIter - 1) / kPerIter;
  const int wave = tid >> 5;
#pragma unroll 1
  for (int it = 0; it < kIters; ++it) {
    const int i  = it * kPerIter + wave * 16 + m;       // node-slice id
    const int ic = (i < kNodeSlices) ? i : 0;           // clamp (EXEC stays full)
    const int s  = ic / kN;
    const int o  = s * kSliceF + 2 * (ic - s * kN);
    const float2 ag = *(const float2*)(s_agg + o);      // ds_load_b64
    const float2 xv = *(const float2*)(s_x + o);        // ds_load_b64
    // B1: K0=agg0,K1=agg1 (lanes 0-15) / K2=x0,K3=x1 (lanes 16-31).
    v2f bm; bm.x = lo ? ag.x : xv.x;
            bm.y = lo ? ag.y : xv.y;
    v8f c = {}; c[0] = lo ? ubb0 : 0.f; c[1] = lo ? ubb1 : 0.f;
    v8f d = wmma_k4(a1, bm, c);                         // h = [agg,x]@W1 + b1
    d[0] = fmaxf(d[0], 0.f); d[1] = fmaxf(d[1], 0.f);   // relu (rows 2..15 = 0)

    bm.x = d[0]; bm.y = d[1];                           // D->B: layout identical
    c[0] = lo ? (xv.x + vb0) : 0.f;                     // bias + residual x
    c[1] = lo ? (xv.y + vb1) : 0.f;
    d = wmma_k4(a2, bm, c);                             // out = h@W2 + b2 + x

    bm.x = d[0]; bm.y = d[1];
    c[0] = lo ? fb10 : 0.f; c[1] = lo ? fb11 : 0.f;
    d = wmma_k4(a3, bm, c);                             // y1 = out@Wf1 + fb1
    d[0] = fmaxf(d[0], 0.f); d[1] = fmaxf(d[1], 0.f);   // relu

    bm.x = d[0]; bm.y = d[1];
    c[0] = lo ? qb : 0.f; c[1] = 0.f;
    d = wmma_k4(a4, bm, c);                             // z = y1@Wf2 + fb2

    const float y = sigm(d[0]);                         // row0, lanes 0-15
    if (lo && i < kNodeSlices) {
      out[(size_t)blockIdx.x * kNodeSlices + i] = y;
    }
  }
}

}  // namespace

extern "C" void kernel_launch(void* const* d_in, const int* in_sizes, int n_in,
                              void* d_out, int out_size, void* d_ws, size_t ws_size,
                              hipStream_t stream) {
  (void)in_sizes; (void)n_in; (void)out_size; (void)d_ws; (void)ws_size;
  const float*     x  = (const float*)d_in[0];
  const long long* ei = (const long long*)d_in[1];  // int64 edge_index
  gnn_fused<<<dim3(kBlocks), dim3(kThreads), 0, stream>>>(
      x, ei,
      (const float*)d_in[2],  (const float*)d_in[3],   // msg_w1, msg_b1
      (const float*)d_in[4],  (const float*)d_in[5],   // msg_w2, msg_b2
      (const float*)d_in[6],  (const float*)d_in[7],   // gate_w, gate_b
      (const float*)d_in[8],  (const float*)d_in[9],   // up_w1,  up_b1
      (const float*)d_in[10], (const float*)d_in[11],  // up_w2,  up_b2
      (const float*)d_in[12], (const float*)d_in[13],  // fc_w1,  fc_b1
      (const float*)d_in[14], (const float*)d_in[15],  // fc_w2,  fc_b2
      (float*)d_out);
}